// Model_33285996544262
// MI455X (gfx1250) — compile-verified
//
#include <hip/hip_runtime.h>
#include <hip/hip_bf16.h>
#include <math.h>

#define DMODEL 768
#define NPKG   12
#define NFUN   32
#define NROWJ  (NPKG + NFUN)          // 44 rows per batch item
#define BATCH  64
#define NROWS  (BATCH * NROWJ)        // 2816 total rows (176 M-tiles of 16)
#define MT     (NROWS / 16)           // 176
#define NT     (DMODEL / 16)          // 48

typedef __attribute__((ext_vector_type(2))) float v2f;
typedef __attribute__((ext_vector_type(8))) float v8f;

// ---------------------------------------------------------------------------
// Kernel A: T[b, j, :] = emb[b, j, :] @ han_w.T + han_b  via fp32 WMMA.
// One wave computes one 16x16 output tile; K loop steps by 4 (16x16x4 f32).
// A-frag layout (32-bit A 16x4): lanes 0-15 -> K = {k, k+1}, lanes 16-31 ->
// K = {k+2, k+3}, M = lane&15.  B-frag mirrors it with N = lane&15.
// ---------------------------------------------------------------------------
__global__ __launch_bounds__(128) void han_transform_gemm(
    const float* __restrict__ pkg,   // (64, 12, 768)
    const float* __restrict__ fun,   // (64, 32, 768)
    const float* __restrict__ W,     // (768, 768)  row o = han_w[o, :]
    const float* __restrict__ hb,    // (768,)
    float* __restrict__ T)           // ws: (64, 44, 768)
{
    const int wave = threadIdx.x >> 5;
    const int lane = threadIdx.x & 31;
    const int tile = blockIdx.x * 4 + wave;
    if (tile >= MT * NT) return;

    const int mt   = tile / NT;
    const int nt   = tile % NT;
    const int half = lane >> 4;       // 0 or 1
    const int l16  = lane & 15;

    // A row handled by this lane (M = l16 within the tile)
    const int row = mt * 16 + l16;
    const int b   = row / NROWJ;
    const int j   = row % NROWJ;
    const float* arow = (j < NPKG)
        ? (pkg + ((size_t)b * NPKG + j) * DMODEL)
        : (fun + ((size_t)b * NFUN + (j - NPKG)) * DMODEL);

    // B column (output dim) handled by this lane: B[k][n] = W[n][k]
    const int n = nt * 16 + l16;
    const float* wrow = W + (size_t)n * DMODEL;

    v8f acc = {};
    #pragma unroll 4
    for (int k = 0; k < DMODEL; k += 4) {
        const int ka = k + half * 2;
        v2f a  = *(const v2f*)(arow + ka);
        v2f bb = *(const v2f*)(wrow + ka);
        acc = __builtin_amdgcn_wmma_f32_16x16x4_f32(
            /*neg_a=*/false, a, /*neg_b=*/false, bb,
            /*c_mod=*/(short)0, acc, /*reuse_a=*/false, /*reuse_b=*/false);
    }

    const float bias = hb[n];
    // D layout: VGPR g -> M = g (lanes 0-15) / g+8 (lanes 16-31), N = lane&15
    float* outp = T + (size_t)(mt * 16) * DMODEL + nt * 16;
    #pragma unroll
    for (int g = 0; g < 8; ++g) {
        const int m = g + half * 8;
        outp[(size_t)m * DMODEL + l16] = acc[g] + bias;
    }
}

// ---------------------------------------------------------------------------
// Block-wide HAN attention: att[i] = dot(tanh(rowp[i]), q); softmax over i;
// dst[d] = sum_i w[i] * rowp[i][d].  rowp entries are generic pointers
// (global T rows or LDS temporaries).  blockDim.x == 256 (8 waves).
// ---------------------------------------------------------------------------
__device__ __forceinline__ void block_han(
    const float* const* rowp, int n, const float* q,
    float* att, float* wgt, float* dst,
    int tid, int wave, int lane)
{
    for (int base = 0; base < n; base += 8) {
        const int i = base + wave;
        if (i < n) {
            const float* r = rowp[i];
            float s = 0.f;
            for (int d = lane; d < DMODEL; d += 32)
                s += tanhf(r[d]) * q[d];
            #pragma unroll
            for (int off = 16; off > 0; off >>= 1)
                s += __shfl_down(s, off, 32);
            if (lane == 0) att[i] = s;
        }
    }
    __syncthreads();
    if (tid == 0) {
        float m = att[0];
        for (int i = 1; i < n; ++i) m = fmaxf(m, att[i]);
        float sum = 0.f;
        for (int i = 0; i < n; ++i) { float e = expf(att[i] - m); wgt[i] = e; sum += e; }
        const float inv = 1.f / sum;
        for (int i = 0; i < n; ++i) wgt[i] *= inv;
    }
    __syncthreads();
    for (int d = tid; d < DMODEL; d += 256) {
        float s = 0.f;
        for (int i = 0; i < n; ++i) s += wgt[i] * rowp[i][d];
        dst[d] = s;
    }
    __syncthreads();
}

// ---------------------------------------------------------------------------
// Kernel B: one block per batch item.  Rebuilds the tree exactly like the
// Python state machine, reduces it bottom-up with block_han, then applies the
// final 2-way fusion softmax.
// ---------------------------------------------------------------------------
__global__ __launch_bounds__(256) void tree_fuse(
    const float* __restrict__ nl,      // (64, 768)
    const int*   __restrict__ levels,  // (64, 32)
    const float* __restrict__ fuw,     // (1, 768)
    const float* __restrict__ fub,     // (1,)
    float* __restrict__ T,             // ws: (64, 44, 768), rows updated in place
    float* __restrict__ out)           // (64, 768)
{
    const int b    = blockIdx.x;
    const int tid  = threadIdx.x;
    const int wave = tid >> 5;
    const int lane = tid & 31;

    __shared__ float q[DMODEL];
    __shared__ float va[DMODEL];
    __shared__ float vb[DMODEL];
    __shared__ float att[48];
    __shared__ float wgt[48];
    __shared__ const float* rowp[48];
    __shared__ int s_n;
    __shared__ int s_par[NFUN];        // -1 root, -2 skipped, >=0 function idx
    __shared__ int s_lev[NFUN];
    __shared__ int s_has[NFUN];

    for (int d = tid; d < DMODEL; d += 256) q[d] = nl[(size_t)b * DMODEL + d];

    if (tid == 0) {
        for (int f = 0; f < NFUN; ++f) {
            s_par[f] = -2;
            s_lev[f] = levels[b * NFUN + f];
        }
        int curLevel = -1;
        int curNode  = -1;             // -1 == root
        for (int f = 0; f < NFUN; ++f) {
            const int l = s_lev[f];
            if (l == -1) break;
            int p;
            if (l - curLevel == 1) {
                p = curNode;
            } else if (l == curLevel && curNode >= 0) {
                p = s_par[curNode];
            } else if (l < curLevel && curNode >= 0) {
                p = s_par[curNode];
                for (int i = 0; i < curLevel - l; ++i)
                    p = (p >= 0) ? s_par[p] : -1;
            } else {
                continue;              // node silently dropped (level jump > 1)
            }
            s_par[f] = p;
            curNode  = f;
            curLevel = l;
        }
        for (int f = 0; f < NFUN; ++f) s_has[f] = 0;
        for (int g = 0; g < NFUN; ++g)
            if (s_par[g] >= 0) s_has[s_par[g]] = 1;
    }
    __syncthreads();

    float* Tb = T + (size_t)b * NROWJ * DMODEL;

    // internal nodes: level 1 first (children are level-2 leaves), then level 0
    for (int pass = 1; pass >= 0; --pass) {
        for (int f = 0; f < NFUN; ++f) {
            if (s_lev[f] != pass) continue;
            if (s_par[f] == -2)   continue;   // not attached
            if (!s_has[f])        continue;   // leaf: ctx == transformed row
            if (tid == 0) {
                int n = 0;
                for (int g = 0; g < NFUN; ++g)
                    if (s_par[g] == f) rowp[n++] = Tb + (size_t)(NPKG + g) * DMODEL;
                s_n = n;
            }
            __syncthreads();
            block_han(rowp, s_n, q, att, wgt, va, tid, wave, lane);   // child
            if (tid == 0) {
                rowp[0] = Tb + (size_t)(NPKG + f) * DMODEL;           // cur
                rowp[1] = va;
                s_n = 2;
            }
            __syncthreads();
            block_han(rowp, 2, q, att, wgt, vb, tid, wave, lane);
            for (int d = tid; d < DMODEL; d += 256)
                Tb[(size_t)(NPKG + f) * DMODEL + d] = vb[d];
            __threadfence_block();
            __syncthreads();
        }
    }

    // root: 12 packages + attached level-0 functions
    if (tid == 0) {
        int n = 0;
        for (int j = 0; j < NPKG; ++j) rowp[n++] = Tb + (size_t)j * DMODEL;
        for (int f = 0; f < NFUN; ++f)
            if (s_par[f] == -1) rowp[n++] = Tb + (size_t)(NPKG + f) * DMODEL;
        s_n = n;
    }
    __syncthreads();
    block_han(rowp, s_n, q, att, wgt, va, tid, wave, lane);   // va = ctx_b

    // fusion: 2-way softmax over [dot(nl, fuw)+fub, dot(ctx, fuw)+fub]
    if (wave < 2) {
        const float* v = (wave == 0) ? q : va;
        float s = 0.f;
        for (int d = lane; d < DMODEL; d += 32) s += v[d] * fuw[d];
        #pragma unroll
        for (int off = 16; off > 0; off >>= 1) s += __shfl_down(s, off, 32);
        if (lane == 0) att[wave] = s + fub[0];
    }
    __syncthreads();
    if (tid == 0) {
        const float m  = fmaxf(att[0], att[1]);
        const float e0 = expf(att[0] - m);
        const float e1 = expf(att[1] - m);
        const float inv = 1.f / (e0 + e1);
        wgt[0] = e0 * inv;
        wgt[1] = e1 * inv;
    }
    __syncthreads();
    for (int d = tid; d < DMODEL; d += 256)
        out[(size_t)b * DMODEL + d] = wgt[0] * q[d] + wgt[1] * va[d];
}

extern "C" void kernel_launch(void* const* d_in, const int* in_sizes, int n_in,
                              void* d_out, int out_size, void* d_ws, size_t ws_size,
                              hipStream_t stream) {
    const float* nl  = (const float*)d_in[0];   // nl_vec        (64, 768)
    const float* pkg = (const float*)d_in[1];   // package_embs  (64, 12, 768)
    const float* fun = (const float*)d_in[2];   // function_embs (64, 32, 768)
    const int*   lev = (const int*)  d_in[3];   // levels        (64, 32)
    const float* hw  = (const float*)d_in[4];   // han_w         (768, 768)
    const float* hb  = (const float*)d_in[5];   // han_b         (768,)
    const float* fw  = (const float*)d_in[6];   // fu_w          (1, 768)
    const float* fb  = (const float*)d_in[7];   // fu_b          (1,)
    float* out = (float*)d_out;                 // (64, 768)
    float* T   = (float*)d_ws;                  // (64, 44, 768) = 8.65 MB

    // 176 * 48 = 8448 tiles, 4 waves (tiles) per 128-thread block
    han_transform_gemm<<<(MT * NT) / 4, 128, 0, stream>>>(pkg, fun, hw, hb, T);
    tree_fuse<<<BATCH, 256, 0, stream>>>(nl, lev, fw, fb, T, out);
}